// PNCRC_38388417692206
// MI455X (gfx1250) — compile-verified
//
#include <hip/hip_runtime.h>
#include <math.h>

typedef __attribute__((ext_vector_type(2))) float v2f;
typedef __attribute__((ext_vector_type(8))) float v8f;

#define BATCH 64
#define HDIM 32
#define WDIM 32
#define KDIM 32
#define FEAT 256
#define NLAYERS 11
#define HW (HDIM * WDIM)

// ---------------------------------------------------------------------------
// 1) Per-(k,r,c) logsumexp over the FEAT axis of leaf -> norm[k*HW + rc]
// ---------------------------------------------------------------------------
__global__ __launch_bounds__(256) void pn_norm_kernel(const float* __restrict__ leaf,
                                                      float* __restrict__ norm) {
    int id = blockIdx.x * blockDim.x + threadIdx.x;   // 32*1024 = 32768
    if (id >= KDIM * HW) return;
    int rc = id & (HW - 1);
    int k  = id >> 10;
    const float* base = leaf + (size_t)k * FEAT * HW + rc;
    float m = -INFINITY;
    for (int f = 0; f < FEAT; ++f) m = fmaxf(m, base[(size_t)f * HW]);
    float s = 0.f;
    for (int f = 0; f < FEAT; ++f) s += __expf(base[(size_t)f * HW] - m);
    norm[id] = m + __logf(s);
}

// ---------------------------------------------------------------------------
// 2) Gather observed pixel log-likelihoods: logp[site][k], site = b*HW + rc
// ---------------------------------------------------------------------------
__global__ __launch_bounds__(256) void pn_gather_kernel(const int* __restrict__ x,
                                                        const float* __restrict__ leaf,
                                                        const float* __restrict__ norm,
                                                        float* __restrict__ out) {
    int s = blockIdx.x * blockDim.x + threadIdx.x;    // 64*1024 sites
    if (s >= BATCH * HW) return;
    int rc = s & (HW - 1);
    int xv = x[s];
    #pragma unroll
    for (int k = 0; k < KDIM; ++k) {
        float v = leaf[(size_t)k * FEAT * HW + (size_t)xv * HW + rc] - norm[k * HW + rc];
        out[(size_t)s * KDIM + k] = v;
    }
}

// ---------------------------------------------------------------------------
// 3) Mixing layer:
//    out[site][p] = lse_q(logp[site][q] + w[p][q]) - lse_q(w[p][q])
//    where w[p][q] = sum_k logp[site][k] * Wi[p][q][k]
//    GEMM via V_WMMA_F32_16X16X4_F32: one wave owns a 16-site M-tile,
//    loops p (32), 2 N-tiles (q 0-15 / 16-31), 8 K-slices of 4.
// ---------------------------------------------------------------------------
__global__ __launch_bounds__(256) void pn_mixing_kernel(const float* __restrict__ in,
                                                        float* __restrict__ out,
                                                        const float* __restrict__ Wi,
                                                        int n_tiles) {
    __shared__ float lp_lds[8][16 * KDIM];   // per-wave 16-site x 32-k staging (16KB)

    const int tid  = threadIdx.x;
    const int lane = tid & 31;
    const int wave = tid >> 5;
    const int m    = lane & 15;              // row-in-tile / q-in-tile
    const int hi   = (lane >> 4) & 1;        // lane half

    int tile = blockIdx.x * 8 + wave;

    // Stage this wave's 16x32 logp tile into LDS (512 floats = 4 float4/lane)
    if (tile < n_tiles) {
        const float* src = in + (size_t)tile * 16 * KDIM;
        float4* dst4 = (float4*)&lp_lds[wave][0];
        const float4* src4 = (const float4*)src;
        for (int j = lane; j < 128; j += 32) dst4[j] = src4[j];
    }
    __syncthreads();                          // all waves reach this
    if (tile >= n_tiles) return;

    const float* lp = &lp_lds[wave][0];

    // A-fragments: lane holds A[m][kt*4 + 2*hi .. +1]  (16x4 f32 layout)
    v2f afrag[8];
    #pragma unroll
    for (int kt = 0; kt < 8; ++kt) {
        int kk = kt * 4 + hi * 2;
        afrag[kt] = *(const v2f*)(lp + m * KDIM + kk);
    }
    // logp values aligned with the C-matrix layout (site = v + 8*hi, q = m / m+16)
    float lv0[8], lv1[8];
    #pragma unroll
    for (int v = 0; v < 8; ++v) {
        int site = v + hi * 8;
        lv0[v] = lp[site * KDIM + m];
        lv1[v] = lp[site * KDIM + m + 16];
    }

    float* dst = out + (size_t)tile * 16 * KDIM;

    #pragma unroll 1
    for (int p = 0; p < 32; ++p) {
        v8f acc0 = {};   // q = 0..15
        v8f acc1 = {};   // q = 16..31
        const float* wp = Wi + (size_t)p * KDIM * KDIM;   // Wi[p][q][k], k contiguous
        #pragma unroll
        for (int kt = 0; kt < 8; ++kt) {
            int kk = kt * 4 + hi * 2;
            // B 4x16 frags: lane holds B[kk..kk+1][q], q = m (+16)
            v2f b0 = *(const v2f*)(wp + (size_t)m * KDIM + kk);
            v2f b1 = *(const v2f*)(wp + (size_t)(m + 16) * KDIM + kk);
            acc0 = __builtin_amdgcn_wmma_f32_16x16x4_f32(false, afrag[kt], false, b0,
                                                         (short)0, acc0, false, false);
            acc1 = __builtin_amdgcn_wmma_f32_16x16x4_f32(false, afrag[kt], false, b1,
                                                         (short)0, acc1, false, false);
        }
        // per site row v: two 32-wide logsumexps across q (lanes within a half)
        #pragma unroll
        for (int v = 0; v < 8; ++v) {
            float u0 = acc0[v], u1 = acc1[v];           // w values
            float t0 = u0 + lv0[v], t1 = u1 + lv1[v];   // w + logp
            float mw = fmaxf(u0, u1);
            float mt = fmaxf(t0, t1);
            #pragma unroll
            for (int d = 1; d < 16; d <<= 1) {
                mw = fmaxf(mw, __shfl_xor(mw, d, 32));
                mt = fmaxf(mt, __shfl_xor(mt, d, 32));
            }
            float sw = __expf(u0 - mw) + __expf(u1 - mw);
            float st = __expf(t0 - mt) + __expf(t1 - mt);
            #pragma unroll
            for (int d = 1; d < 16; d <<= 1) {
                sw += __shfl_xor(sw, d, 32);
                st += __shfl_xor(st, d, 32);
            }
            float res = (mt + __logf(st)) - (mw + __logf(sw));
            if (m == (p & 15)) {                        // one lane per half stores
                int site = v + hi * 8;
                dst[site * KDIM + p] = res;
            }
        }
    }
}

// ---------------------------------------------------------------------------
// 4) Spatial product layers (pairwise adds; all dims here are even)
// ---------------------------------------------------------------------------
__global__ __launch_bounds__(256) void pn_halve_cols(const float* __restrict__ in,
                                                     float* __restrict__ out, int R, int C) {
    int Ch = C >> 1;
    int n = BATCH * R * Ch * KDIM;
    int idx = blockIdx.x * blockDim.x + threadIdx.x;
    if (idx >= n) return;
    int k  = idx & (KDIM - 1);
    int so = idx >> 5;
    int c  = so % Ch;
    int t  = so / Ch;                 // b*R + r
    int si = t * C + 2 * c;
    out[idx] = in[(size_t)si * KDIM + k] + in[(size_t)(si + 1) * KDIM + k];
}

__global__ __launch_bounds__(256) void pn_halve_rows(const float* __restrict__ in,
                                                     float* __restrict__ out, int R, int C) {
    int Rh = R >> 1;
    int n = BATCH * Rh * C * KDIM;
    int idx = blockIdx.x * blockDim.x + threadIdx.x;
    if (idx >= n) return;
    int k  = idx & (KDIM - 1);
    int so = idx >> 5;
    int c  = so % C;
    int t  = so / C;                  // b*Rh + r
    int r  = t % Rh;
    int b  = t / Rh;
    int si0 = (b * R + 2 * r) * C + c;
    int si1 = (b * R + 2 * r + 1) * C + c;
    out[idx] = in[(size_t)si0 * KDIM + k] + in[(size_t)si1 * KDIM + k];
}

// ---------------------------------------------------------------------------
// 5) Root mixture: out[b] = lse_k(logp[b,k] + root_w[k]) - lse_k(root_w[k])
// ---------------------------------------------------------------------------
__global__ void pn_root_kernel(const float* __restrict__ logp,
                               const float* __restrict__ root_w,
                               float* __restrict__ out) {
    int b = threadIdx.x;
    if (b >= BATCH) return;
    float mr = -INFINITY;
    for (int k = 0; k < KDIM; ++k) mr = fmaxf(mr, root_w[k]);
    float sr = 0.f;
    for (int k = 0; k < KDIM; ++k) sr += __expf(root_w[k] - mr);
    float lse_r = mr + __logf(sr);
    float m = -INFINITY;
    for (int k = 0; k < KDIM; ++k) m = fmaxf(m, logp[b * KDIM + k] + root_w[k]);
    float s = 0.f;
    for (int k = 0; k < KDIM; ++k) s += __expf(logp[b * KDIM + k] + root_w[k] - m);
    out[b] = m + __logf(s) - lse_r;
}

// ---------------------------------------------------------------------------
extern "C" void kernel_launch(void* const* d_in, const int* in_sizes, int n_in,
                              void* d_out, int out_size, void* d_ws, size_t ws_size,
                              hipStream_t stream) {
    const int*   x      = (const int*)d_in[0];
    const float* leaf   = (const float*)d_in[1];
    const float* mix_w  = (const float*)d_in[2];
    const float* root_w = (const float*)d_in[3];

    float* bufA = (float*)d_ws;               // 2,097,152 floats (8 MB)
    float* bufB = bufA + 2097152;
    float* bufC = bufB + 2097152;             // mixing output (ping)
    float* norm = bufC + 2097152;             // 32,768 floats

    pn_norm_kernel<<<128, 256, 0, stream>>>(leaf, norm);
    pn_gather_kernel<<<(BATCH * HW + 255) / 256, 256, 0, stream>>>(x, leaf, norm, bufA);

    float* cur = bufA;
    float* nxt = bufB;
    int R = HDIM, C = WDIM;
    for (int i = 0; i < NLAYERS; ++i) {
        int n_sites = BATCH * R * C;
        int tiles   = n_sites / 16;           // always divisible (>= 64 sites)
        int blocks  = (tiles + 7) / 8;
        pn_mixing_kernel<<<blocks, 256, 0, stream>>>(
            cur, bufC, mix_w + (size_t)i * KDIM * KDIM * KDIM, tiles);
        if (i < NLAYERS - 1) {
            if ((i & 1) == 0) {
                int n = BATCH * R * (C / 2) * KDIM;
                pn_halve_cols<<<(n + 255) / 256, 256, 0, stream>>>(bufC, nxt, R, C);
                C >>= 1;
            } else {
                int n = BATCH * (R / 2) * C * KDIM;
                pn_halve_rows<<<(n + 255) / 256, 256, 0, stream>>>(bufC, nxt, R, C);
                R >>= 1;
            }
            float* t = cur; cur = nxt; nxt = t;
        }
    }
    // after layer 10: bufC holds [b][k] (1x1 spatial)
    pn_root_kernel<<<1, 64, 0, stream>>>(bufC, root_w, (float*)d_out);
}